// ComputeTargets_34815004902004
// MI455X (gfx1250) — compile-verified
//
#include <hip/hip_runtime.h>
#include <hip/hip_bf16.h>
#include <cstdint>

#define NGT  64   // GT boxes per image (reference: N=64)
#define NCLS 80   // NUM_CLASSES

typedef float v4f __attribute__((ext_vector_type(4)));

// ---- CDNA5 async global->LDS path (guarded; falls back to plain LDS fill) --
#if defined(__has_builtin)
#  if __has_builtin(__builtin_amdgcn_global_load_async_to_lds_b32) && \
      __has_builtin(__builtin_amdgcn_s_wait_asynccnt)
#    define ASYNC_LDS 1
#  endif
#endif

// clang-22 signature: (int AS1*, int AS3*, imm offset, imm cpol)
typedef __attribute__((address_space(1))) int glb_i32_t;
typedef __attribute__((address_space(3))) int lds_i32_t;

__device__ __forceinline__ glb_i32_t* to_glb(const void* p) {
  return (glb_i32_t*)(uintptr_t)p;  // generic global ptr == global address
}
// Generic LDS pointer low 32 bits == LDS byte offset.
__device__ __forceinline__ lds_i32_t* to_lds(const void* p) {
  return (lds_i32_t*)(uint32_t)(uintptr_t)p;
}

extern "C" __global__ __launch_bounds__(256)
void targets_kernel(const float* __restrict__ ann,      // (B, 64, 5)
                    const float* __restrict__ anchors,  // (A, 4)
                    float* __restrict__ cls_out,        // (B, A, 80)
                    float* __restrict__ reg_out,        // (B, A, 4)
                    float* __restrict__ st_out,         // (B, A)
                    int A)
{
  __shared__ __align__(16) float raw[NGT * 5];
  __shared__ v4f   box[NGT];
  __shared__ float areaB[NGT];
  __shared__ int   lab[NGT];

  const int t = threadIdx.x;
  const int b = blockIdx.y;
  const int a = blockIdx.x * 256 + t;
  const float* g = ann + (size_t)b * (NGT * 5);

  // Overlap this thread's anchor fetch with the LDS staging + barriers.
  if (a < A) __builtin_prefetch(anchors + (size_t)a * 4, 0, 3);

  // ---- Stage per-batch annotations (320 dwords) into LDS ----
#if ASYNC_LDS
  if (t < NGT * 5)
    __builtin_amdgcn_global_load_async_to_lds_b32(to_glb(g + t), to_lds(raw + t), 0, 0);
  {
    int i = t + 256;
    if (i < NGT * 5)
      __builtin_amdgcn_global_load_async_to_lds_b32(to_glb(g + i), to_lds(raw + i), 0, 0);
  }
  __builtin_amdgcn_s_wait_asynccnt(0);
#else
  for (int i = t; i < NGT * 5; i += 256) raw[i] = g[i];
#endif
  __syncthreads();

  // SoA repack: box as float4 (one ds_load_b128 per iteration), area, int label.
  if (t < NGT) {
    float x1 = raw[t * 5 + 0], y1 = raw[t * 5 + 1];
    float x2 = raw[t * 5 + 2], y2 = raw[t * 5 + 3];
    v4f bb; bb.x = x1; bb.y = y1; bb.z = x2; bb.w = y2;
    box[t]   = bb;
    areaB[t] = (x2 - x1) * (y2 - y1);
    lab[t]   = (int)raw[t * 5 + 4];
  }
  __syncthreads();

  if (a >= A) return;

  const v4f an = ((const v4f*)anchors)[a];               // global_load_b128
  const float area_a = (an.z - an.x) * (an.w - an.y);

  // Argmax over 64 boxes without divisions: compare inter/uni ratios by
  // cross-multiplication (both unions > 0). Strict '>' keeps the first
  // maximum, matching jnp.argmax tie-breaking.
  float best_i = -1.0f, best_u = 1.0f;
  int best_j = 0;
#pragma unroll 8
  for (int j = 0; j < NGT; ++j) {
    v4f bb = box[j];
    float iw    = fminf(an.z, bb.z) - fmaxf(an.x, bb.x);
    float ih    = fminf(an.w, bb.w) - fmaxf(an.y, bb.y);
    float inter = fmaxf(iw, 0.0f) * fmaxf(ih, 0.0f);
    float uni   = fmaxf(area_a + areaB[j] - inter, 1e-8f);
    bool better = inter * best_u > best_i * uni;
    best_i = better ? inter : best_i;
    best_u = better ? uni   : best_u;
    best_j = better ? j     : best_j;
  }

  // Thresholds on max_overlap = best_i / best_u, division-free.
  const bool  positive = best_i >= 0.5f * best_u;
  const bool  ignore   = (best_i > 0.4f * best_u) && !positive;
  const float state    = positive ? 1.0f : (ignore ? -1.0f : 0.0f);

  const size_t row = (size_t)b * (size_t)A + (size_t)a;

  // ---- cls one-hot: 20 x b128 NT zero stores + single hot overwrite ----
  float* cls  = cls_out + row * NCLS;
  v4f*   cls4 = (v4f*)cls;
  v4f z = {0.0f, 0.0f, 0.0f, 0.0f};
#pragma unroll
  for (int k = 0; k < NCLS / 4; ++k)
    __builtin_nontemporal_store(z, cls4 + k);
  if (positive)
    __builtin_nontemporal_store(1.0f, cls + lab[best_j]);  // same-thread stores ordered

  // ---- regression target: ((box - anchor)/wh)/0.2 ----
  v4f bb = box[best_j];
  const float aw = an.z - an.x, ah = an.w - an.y;
  const float iaw = 5.0f / aw, iah = 5.0f / ah;
  v4f r;
  r.x = (bb.x - an.x) * iaw;
  r.y = (bb.y - an.y) * iah;
  r.z = (bb.z - an.z) * iaw;
  r.w = (bb.w - an.w) * iah;
  __builtin_nontemporal_store(r, (v4f*)reg_out + row);

  __builtin_nontemporal_store(state, st_out + row);
}

extern "C" void kernel_launch(void* const* d_in, const int* in_sizes, int n_in,
                              void* d_out, int out_size, void* d_ws, size_t ws_size,
                              hipStream_t stream) {
  const float* ann     = (const float*)d_in[0];  // (B, 64, 5) float32
  const float* anchors = (const float*)d_in[1];  // (A, 4)    float32
  const int B = in_sizes[0] / (NGT * 5);
  const int A = in_sizes[1] / 4;

  float* out = (float*)d_out;                    // cls | reg | states, concatenated
  float* cls = out;
  float* reg = cls + (size_t)B * (size_t)A * NCLS;
  float* st  = reg + (size_t)B * (size_t)A * 4;

  dim3 block(256, 1, 1);
  dim3 grid((A + 255) / 256, B, 1);
  hipLaunchKernelGGL(targets_kernel, grid, block, 0, stream,
                     ann, anchors, cls, reg, st, A);
}